// ScaledDotProductAttention_24309514895892
// MI455X (gfx1250) — compile-verified
//
#include <hip/hip_runtime.h>

typedef __attribute__((ext_vector_type(2))) float v2f;
typedef __attribute__((ext_vector_type(8))) float v8f;

#define S_LEN 2048
#define D_DIM 64
#define NKT   (S_LEN / 16)   // 128 key tiles
#define LDSTR 20             // padded LDS row stride (floats): conflict-free + 8B aligned

__device__ __forceinline__ float rowmax16(float t) {
  t = fmaxf(t, __shfl_xor(t, 1, 32));
  t = fmaxf(t, __shfl_xor(t, 2, 32));
  t = fmaxf(t, __shfl_xor(t, 4, 32));
  t = fmaxf(t, __shfl_xor(t, 8, 32));
  return t;
}
__device__ __forceinline__ float rowsum16(float t) {
  t += __shfl_xor(t, 1, 32);
  t += __shfl_xor(t, 2, 32);
  t += __shfl_xor(t, 4, 32);
  t += __shfl_xor(t, 8, 32);
  return t;
}

// QK^T 16x16 score tile: two independent WMMA accumulator chains (even/odd
// K-groups) so the scheduler can interleave them, then one vector add.
__device__ __forceinline__ v8f qk_tile(const v2f* __restrict__ qa,
                                       const float* __restrict__ Kt) {
  v8f c0 = {0.f, 0.f, 0.f, 0.f, 0.f, 0.f, 0.f, 0.f};
  v8f c1 = {0.f, 0.f, 0.f, 0.f, 0.f, 0.f, 0.f, 0.f};
  #pragma unroll
  for (int g = 0; g < 16; g += 2) {
    v2f b0 = *(const v2f*)(Kt + 4 * g);
    v2f b1 = *(const v2f*)(Kt + 4 * (g + 1));
    c0 = __builtin_amdgcn_wmma_f32_16x16x4_f32(false, qa[g], false, b0,
                                               (short)0, c0, false, false);
    c1 = __builtin_amdgcn_wmma_f32_16x16x4_f32(false, qa[g + 1], false, b1,
                                               (short)0, c1, false, false);
  }
  return c0 + c1;
}

__global__ __launch_bounds__(128) void sdpa_f32_wmma_kernel(
    const float* __restrict__ Q, const float* __restrict__ K,
    const float* __restrict__ V, const unsigned char* __restrict__ M,
    float* __restrict__ O, float* __restrict__ W)
{
  __shared__ float lds[4][16 * LDSTR];

  const int tid  = threadIdx.x;
  const int wid  = tid >> 5;
  const int lane = tid & 31;
  const int half = lane >> 4;   // 0: lanes 0-15, 1: lanes 16-31
  const int l16  = lane & 15;

  const int gid = blockIdx.x * 4 + wid;   // 0..4095 q-tiles
  const int bh  = gid >> 7;               // (b*H + h)
  const int qt  = gid & 127;
  const int b   = bh >> 4;
  const int q0  = qt * 16;

  const float scale  = 0.125f;                       // 1/sqrt(64)
  const float NEGMIN = -3.4028234663852886e38f;      // finfo(f32).min

  const float* Qb = Q + (size_t)bh * S_LEN * D_DIM;
  const float* Kb = K + (size_t)bh * S_LEN * D_DIM;
  const float* Vb = V + (size_t)bh * S_LEN * D_DIM;
  const unsigned char* Mb = M + (size_t)b * S_LEN * S_LEN + (size_t)q0 * S_LEN;
  float* Ob = O + (size_t)bh * S_LEN * D_DIM;
  float* Wb = W + (size_t)bh * S_LEN * S_LEN + (size_t)q0 * S_LEN;
  float* wlds = lds[wid];

  // ---- Q tile in 32-bit A-operand layout (16x4 per WMMA, 16 K-groups for D=64)
  // lane<16: row=l16, k = 4g+{0,1};  lane>=16: row=l16, k = 4g+{2,3}
  v2f qa[16];
  #pragma unroll
  for (int g = 0; g < 16; ++g)
    qa[g] = *(const v2f*)(Qb + (size_t)(q0 + l16) * D_DIM + 4 * g + 2 * half);

  // per-lane row stats: row r + 8*half (matches C-matrix layout)
  float mrow[8], lrow[8];
  #pragma unroll
  for (int r = 0; r < 8; ++r) { mrow[r] = -__builtin_inff(); lrow[r] = 0.0f; }

  // ================= Pass A: online softmax stats (max, sum) =================
  for (int kt = 0; kt < NKT; ++kt) {
    // B operand = K^T: lane<16: B[k=4g+{0,1}][n=l16] = K[kt*16+l16][4g+{0,1}]
    const float* Kt = Kb + (size_t)(kt * 16 + l16) * D_DIM + 2 * half;
    v8f c = qk_tile(qa, Kt);
    #pragma unroll
    for (int r = 0; r < 8; ++r) {
      const int row = r + 8 * half;
      float s = c[r] * scale;
      if (!Mb[(size_t)row * S_LEN + kt * 16 + l16]) s = NEGMIN;
      float mn = fmaxf(mrow[r], rowmax16(s));
      float p  = __expf(s - mn);
      float ps = rowsum16(p);
      lrow[r]  = lrow[r] * __expf(mrow[r] - mn) + ps;
      mrow[r]  = mn;
    }
  }
  float inv_l[8];
  #pragma unroll
  for (int r = 0; r < 8; ++r) inv_l[r] = 1.0f / lrow[r];

  // ====== Pass B: recompute scores, emit weights, accumulate O = P @ V ======
  v8f acc[4];
  #pragma unroll
  for (int d = 0; d < 4; ++d) acc[d] = (v8f){0.f,0.f,0.f,0.f,0.f,0.f,0.f,0.f};

  for (int kt = 0; kt < NKT; ++kt) {
    const float* Kt = Kb + (size_t)(kt * 16 + l16) * D_DIM + 2 * half;
    v8f c = qk_tile(qa, Kt);

    // normalize, store weights (coalesced 64B per half-wave), stage P in LDS (C layout)
    #pragma unroll
    for (int r = 0; r < 8; ++r) {
      const int row = r + 8 * half;
      float s = c[r] * scale;
      if (!Mb[(size_t)row * S_LEN + kt * 16 + l16]) s = NEGMIN;
      float p = __expf(s - mrow[r]) * inv_l[r];
      Wb[(size_t)row * S_LEN + kt * 16 + l16] = p;
      wlds[row * LDSTR + l16] = p;
    }
    __syncthreads();   // order LDS store -> transposed load (uniform across waves)

    // P @ V: read P back in A-operand layout, V streamed as B operand;
    // 4 independent accumulators (dvt) keep the WMMA pipe busy.
    const float* Vt = Vb + (size_t)(kt * 16 + 2 * half) * D_DIM + l16;
    #pragma unroll
    for (int g = 0; g < 4; ++g) {
      v2f pa = *(const v2f*)(&wlds[l16 * LDSTR + 4 * g + 2 * half]);
      #pragma unroll
      for (int dvt = 0; dvt < 4; ++dvt) {
        v2f vb;
        vb.x = Vt[(size_t)(4 * g)     * D_DIM + dvt * 16];
        vb.y = Vt[(size_t)(4 * g + 1) * D_DIM + dvt * 16];
        acc[dvt] = __builtin_amdgcn_wmma_f32_16x16x4_f32(false, pa, false, vb,
                                                         (short)0, acc[dvt], false, false);
      }
    }
    __syncthreads();   // protect LDS tile before next iteration overwrites
  }

  // ---- store O tile (C layout -> row-major [.., q, d])
  #pragma unroll
  for (int dvt = 0; dvt < 4; ++dvt) {
    #pragma unroll
    for (int r = 0; r < 8; ++r) {
      Ob[(size_t)(q0 + r + 8 * half) * D_DIM + dvt * 16 + l16] = acc[dvt][r];
    }
  }
}

extern "C" void kernel_launch(void* const* d_in, const int* in_sizes, int n_in,
                              void* d_out, int out_size, void* d_ws, size_t ws_size,
                              hipStream_t stream) {
  (void)in_sizes; (void)n_in; (void)out_size; (void)d_ws; (void)ws_size;
  const float* q = (const float*)d_in[0];
  const float* k = (const float*)d_in[1];
  const float* v = (const float*)d_in[2];
  const unsigned char* m = (const unsigned char*)d_in[3];  // bool mask [B,1,S,S]

  float* out = (float*)d_out;                              // [B,H,S,D]
  float* wts = out + (size_t)2 * 16 * S_LEN * D_DIM;       // [B,H,S,S]

  dim3 grid(1024);   // 4096 q-tiles / 4 waves per block
  dim3 block(128);   // 4 wave32 per block
  hipLaunchKernelGGL(sdpa_f32_wmma_kernel, grid, block, 0, stream,
                     q, k, v, m, out, wts);
}